// CellList_2156073583105
// MI455X (gfx1250) — compile-verified
//
#include <hip/hip_runtime.h>
#include <hip/hip_bf16.h>

// Pairwise cell-list screening for N=6144 atoms.
// Outputs (concatenated float32 in d_out):
//   [0,   P)  : atom_pairs row 0 (i indices as float)
//   [P,  2P)  : atom_pairs row 1 (j indices as float)
//   [2P, 3P)  : in_cutoff (0.0 / 1.0)
//   [3P, 6P)  : diff (dx,dy,dz per pair, zeroed when out of cutoff)
//   [6P, 7P)  : dist (0 when out of cutoff)
// P = N(N-1)/2 = 18,871,296  ->  7P floats ~ 528 MB  (store-bandwidth bound)
//
// One wave32 per 16x16 (i,j) tile of the upper triangle.
// d^2 = |ci|^2 + |cj|^2 - 2*ci.cj ; the cross-term Gram tile is one
// V_WMMA_F32_16X16X4_F32 (K = x,y,z,0). diff is computed directly (it is an
// output anyway); coords are 72 KB and stay hot in WGP$/L2.
// All pair indexing is 32-bit (P < 2^31); sqrt is raw v_sqrt_f32.

typedef __attribute__((ext_vector_type(2))) float v2f;
typedef __attribute__((ext_vector_type(8))) float v8f;

#define NATOMS  6144
#define NB      (NATOMS / 16)       // 384 tile rows/cols
#define CUTOFF  5.2f
#define CUTOFF2 (CUTOFF * CUTOFF)

__global__ __launch_bounds__(256) void celllist_wmma_tiles(
    const float* __restrict__ coords,   // [N*3] xyz interleaved
    const int*   __restrict__ species,  // [N]
    float*       __restrict__ out,      // [7P]
    int P)
{
    const int wave = threadIdx.x >> 5;
    const int lane = threadIdx.x & 31;

    const int T = NB * (NB + 1) / 2;                    // 73,920 tiles
    int t = (int)blockIdx.x * 8 + wave;
    if (t >= T) return;                                 // uniform per wave

    // ---- invert triangular tile index: base(ti) = ti*NB - ti*(ti-1)/2 ----
    const float c1 = 2.0f * (float)NB + 1.0f;
    float disc = c1 * c1 - 8.0f * (float)t;
    disc = disc < 0.0f ? 0.0f : disc;
    int ti = (int)((c1 - __builtin_amdgcn_sqrtf(disc)) * 0.5f);
    if (ti < 0) ti = 0;
    if (ti > NB - 1) ti = NB - 1;
    while ((ti + 1) * NB - (ti + 1) * ti / 2 <= t) ++ti;
    while (ti * NB - ti * (ti - 1) / 2 > t)        --ti;
    const int baseti = ti * NB - ti * (ti - 1) / 2;
    const int tj = ti + (t - baseti);                   // tj >= ti

    const int i0 = ti << 4;
    const int j0 = tj << 4;

    const int n  = lane & 15;    // column within tile (j - j0)
    const int hi = lane >> 4;    // half-wave selector

    // ---- per-lane j-column data ----
    const int j = j0 + n;
    const float cjx = coords[3 * j + 0];
    const float cjy = coords[3 * j + 1];
    const float cjz = coords[3 * j + 2];
    const bool  vj  = species[j] != -1;
    const float rj2 = cjx * cjx + cjy * cjy + cjz * cjz;

    // ---- WMMA operands (16x16x4 f32) ----
    // A (16x4, M x K): lanes 0-15 -> K=0,1 (x,y); lanes 16-31 -> K=2,3 (z,0)
    const int ia = i0 + n;
    const float ax = coords[3 * ia + 0];
    const float ay = coords[3 * ia + 1];
    const float az = coords[3 * ia + 2];
    v2f A;
    A.x = hi ? az   : ax;
    A.y = hi ? 0.0f : ay;
    // B (4x16, K x N): mirrored layout, column n per lane
    v2f B;
    B.x = hi ? cjz  : cjx;
    B.y = hi ? 0.0f : cjy;

    v8f dots = {};   // C = 0
    // D = A x B : dots[r] = ci(i0 + 8*hi + r) . cj(j0 + n)
    dots = __builtin_amdgcn_wmma_f32_16x16x4_f32(
        /*neg_a=*/false, A, /*neg_b=*/false, B,
        /*c_mod=*/(short)0, dots, /*reuse_a=*/false, /*reuse_b=*/false);

    // ---- i-row data for this lane's 8 result rows ----
    const int ib = i0 + (hi << 3);
    float cix[8], ciy[8], ciz[8], ri2[8];
    bool  vi[8];
#pragma unroll
    for (int r = 0; r < 8; ++r) {
        const int i = ib + r;
        cix[r] = coords[3 * i + 0];
        ciy[r] = coords[3 * i + 1];
        ciz[r] = coords[3 * i + 2];
        ri2[r] = cix[r] * cix[r] + ciy[r] * ciy[r] + ciz[r] * ciz[r];
        vi[r]  = species[i] != -1;
    }

    float* __restrict__ outPi = out;               // atom_pairs[0]
    float* __restrict__ outPj = out + (size_t)P;   // atom_pairs[1]
    float* __restrict__ outC  = out + (size_t)2 * P;  // in_cutoff
    float* __restrict__ outD  = out + (size_t)3 * P;  // diff (3P)
    float* __restrict__ outR  = out + (size_t)6 * P;  // dist

    // rowbase(ib) once; rowbase(i+1) = rowbase(i) + (N-1-i)  (all 32-bit)
    int rowbase = ib * (NATOMS - 1) - ib * (ib - 1) / 2;

#pragma unroll
    for (int r = 0; r < 8; ++r) {
        const int i = ib + r;
        if (j > i) {                               // tile intersects triu
            const int p = rowbase + (j - i - 1);

            const float dx = cix[r] - cjx;
            const float dy = ciy[r] - cjy;
            const float dz = ciz[r] - cjz;

            float d2 = ri2[r] + rj2 - 2.0f * dots[r];
            d2 = d2 < 0.0f ? 0.0f : d2;

            const bool  incut = (d2 <= CUTOFF2) && vi[r] && vj;
            const float m = incut ? 1.0f : 0.0f;

            outPi[p] = (float)i;
            outPj[p] = (float)j;
            outC[p]  = m;
            outD[3 * p + 0] = dx * m;
            outD[3 * p + 1] = dy * m;
            outD[3 * p + 2] = dz * m;
            outR[p]  = incut ? __builtin_amdgcn_sqrtf(d2) : 0.0f;
        }
        rowbase += (NATOMS - 1) - i;               // advance to next row base
    }
}

extern "C" void kernel_launch(void* const* d_in, const int* in_sizes, int n_in,
                              void* d_out, int out_size, void* d_ws, size_t ws_size,
                              hipStream_t stream) {
    const int*   species = (const int*)d_in[0];    // setup_inputs order: species first
    const float* coords  = (const float*)d_in[1];  // [1, N, 3] float32
    float*       out     = (float*)d_out;

    const int N = NATOMS;
    const int P = N * (N - 1) / 2;                 // 18,871,296 < 2^31

    const int T = NB * (NB + 1) / 2;               // 73,920 tiles
    const int blocks = (T + 7) / 8;                // 8 waves (tiles) / block
    celllist_wmma_tiles<<<blocks, 256, 0, stream>>>(coords, species, out, P);
}